// EpipolarWarpOperator_7129645711513
// MI455X (gfx1250) — compile-verified
//
#include <hip/hip_runtime.h>
#include <math.h>

// Problem constants (match reference)
#define BB 8
#define CC 320
#define HH 64
#define WW 64
#define NSAMP 3

#define CI_CHUNK 80                  // ci channels staged in LDS per step
#define TILE_ELEMS (CI_CHUNK * 54)   // 54 = 3 rows * 18 cols (zero-padded patch)

typedef float v2f __attribute__((ext_vector_type(2)));
typedef float v8f __attribute__((ext_vector_type(8)));

// ---------------------------------------------------------------------------
// Kernel 1: per-batch fundamental matrix. Stores G = F^T (row-major, 9 floats
// per batch) so the epipolar line is  line = G * [px, py, 1]^T.
// ---------------------------------------------------------------------------
__device__ __forceinline__ void rodrigues3(float r0, float r1, float r2, float R[3][3]) {
    float theta = sqrtf(r0 * r0 + r1 * r1 + r2 * r2);
    float inv = 1.0f / fmaxf(theta, 1e-12f);
    float rx = r0 * inv, ry = r1 * inv, rz = r2 * inv;
    float ct = cosf(theta), st = sinf(theta), omc = 1.0f - ct;
    R[0][0] = ct + omc * rx * rx;       R[0][1] = omc * rx * ry - st * rz;  R[0][2] = omc * rx * rz + st * ry;
    R[1][0] = omc * ry * rx + st * rz;  R[1][1] = ct + omc * ry * ry;       R[1][2] = omc * ry * rz - st * rx;
    R[2][0] = omc * rz * rx - st * ry;  R[2][1] = omc * rz * ry + st * rx;  R[2][2] = ct + omc * rz * rz;
    if (theta < 1e-6f) {
        R[0][0] = 1.f; R[0][1] = 0.f; R[0][2] = 0.f;
        R[1][0] = 0.f; R[1][1] = 1.f; R[1][2] = 0.f;
        R[2][0] = 0.f; R[2][1] = 0.f; R[2][2] = 1.f;
    }
}

__device__ __forceinline__ void inv3(const float* K, float I[3][3]) {
    float a = K[0], b = K[1], c = K[2];
    float d = K[3], e = K[4], f = K[5];
    float g = K[6], h = K[7], i = K[8];
    float A =  (e * i - f * h), Bc = -(d * i - f * g), Cc =  (d * h - e * g);
    float det = a * A + b * Bc + c * Cc;
    float id = 1.0f / det;
    I[0][0] = A * id;                  I[0][1] = -(b * i - c * h) * id;   I[0][2] =  (b * f - c * e) * id;
    I[1][0] = Bc * id;                 I[1][1] =  (a * i - c * g) * id;   I[1][2] = -(a * f - c * d) * id;
    I[2][0] = Cc * id;                 I[2][1] = -(a * h - b * g) * id;   I[2][2] =  (a * e - b * d) * id;
}

__global__ void fmat_kernel(const float* __restrict__ Ks, const float* __restrict__ Kt,
                            const float* __restrict__ ps, const float* __restrict__ pt,
                            float* __restrict__ G) {
    int b = threadIdx.x;
    if (b >= BB) return;
    float Rs[3][3], Rt[3][3];
    rodrigues3(ps[b * 6 + 0], ps[b * 6 + 1], ps[b * 6 + 2], Rs);
    rodrigues3(pt[b * 6 + 0], pt[b * 6 + 1], pt[b * 6 + 2], Rt);
    float ts[3] = {ps[b * 6 + 3], ps[b * 6 + 4], ps[b * 6 + 5]};
    float tt[3] = {pt[b * 6 + 3], pt[b * 6 + 4], pt[b * 6 + 5]};
    float Rr[3][3];
    for (int i = 0; i < 3; ++i)
        for (int j = 0; j < 3; ++j)
            Rr[i][j] = Rs[i][0] * Rt[j][0] + Rs[i][1] * Rt[j][1] + Rs[i][2] * Rt[j][2];
    float tr[3];
    for (int i = 0; i < 3; ++i)
        tr[i] = ts[i] - (Rr[i][0] * tt[0] + Rr[i][1] * tt[1] + Rr[i][2] * tt[2]);
    float Sk[3][3] = {{0.f, -tr[2], tr[1]}, {tr[2], 0.f, -tr[0]}, {-tr[1], tr[0], 0.f}};
    float E[3][3];
    for (int i = 0; i < 3; ++i)
        for (int j = 0; j < 3; ++j)
            E[i][j] = Sk[i][0] * Rr[0][j] + Sk[i][1] * Rr[1][j] + Sk[i][2] * Rr[2][j];
    float iKs[3][3], iKt[3][3];
    inv3(Ks + b * 9, iKs);
    inv3(Kt + b * 9, iKt);
    float M1[3][3], F[3][3];
    for (int i = 0; i < 3; ++i)
        for (int k = 0; k < 3; ++k)
            M1[i][k] = iKt[0][i] * E[0][k] + iKt[1][i] * E[1][k] + iKt[2][i] * E[2][k];
    for (int i = 0; i < 3; ++i)
        for (int l = 0; l < 3; ++l)
            F[i][l] = M1[i][0] * iKs[0][l] + M1[i][1] * iKs[1][l] + M1[i][2] * iKs[2][l];
    for (int i = 0; i < 3; ++i)
        for (int j = 0; j < 3; ++j)
            G[b * 9 + i * 3 + j] = F[j][i];
}

// ---------------------------------------------------------------------------
// Kernel 2: epipolar sampling.  One thread = one pixel of one (batch,
// 32-channel group).  Geometry computed once per thread, then the channel
// loop does 12 gathers per channel (L2-resident: x is 42MB < 192MB L2).
// Writes sampled[B, C, H*W] (f32) into workspace, coalesced over pixels.
// ---------------------------------------------------------------------------
#define CH_PER_BLK 32

__global__ __launch_bounds__(256) void sample_kernel(const float* __restrict__ x,
                                                     const float* __restrict__ G,
                                                     float* __restrict__ S) {
    int k = blockIdx.x * blockDim.x + threadIdx.x;   // pixel index 0..4095
    int b = blockIdx.z;
    int c0 = blockIdx.y * CH_PER_BLK;

    const float* g = G + b * 9;
    float px = (float)(k & (WW - 1));
    float py = (float)(k >> 6);
    float la = g[0] * px + g[1] * py + g[2];
    float lb = g[3] * px + g[4] * py + g[5];
    float lc = g[6] * px + g[7] * py + g[8];

    const float W1 = (float)(WW - 1), H1 = (float)(HH - 1), EPS = 1e-10f;
    float ia = 1.0f / (la + EPS);
    float ib = 1.0f / (lb + EPS);
    float x1 = fminf(fmaxf(-lc * ia, 0.0f), W1);
    float x2 = fminf(fmaxf(-(lb * H1 + lc) * ia, 0.0f), W1);
    float y1 = fminf(fmaxf(-lc * ib, 0.0f), H1);
    float y2 = fminf(fmaxf(-(la * W1 + lc) * ib, 0.0f), H1);

    int   idx[NSAMP][4];
    float wgt[NSAMP][4];
#pragma unroll
    for (int s = 0; s < NSAMP; ++s) {
        float t = (float)s * (1.0f / (float)(NSAMP - 1));
        float sx = x1 * (1.0f - t) + x2 * t;
        float sy = y1 * (1.0f - t) + y2 * t;
        float fx0 = floorf(sx), fy0 = floorf(sy);
        float wx = sx - fx0, wy = sy - fy0;
        int x0i = (int)fminf(fmaxf(fx0, 0.0f), W1);
        int x1i = (int)fminf(fmaxf(fx0 + 1.0f, 0.0f), W1);
        int y0i = (int)fminf(fmaxf(fy0, 0.0f), H1);
        int y1i = (int)fminf(fmaxf(fy0 + 1.0f, 0.0f), H1);
        idx[s][0] = y0i * WW + x0i;
        idx[s][1] = y0i * WW + x1i;
        idx[s][2] = y1i * WW + x0i;
        idx[s][3] = y1i * WW + x1i;
        wgt[s][0] = (1.0f - wx) * (1.0f - wy);
        wgt[s][1] = wx * (1.0f - wy);
        wgt[s][2] = (1.0f - wx) * wy;
        wgt[s][3] = wx * wy;
    }

    const float* xb = x + ((size_t)b * CC + c0) * (HH * WW);
    float*       sb = S + ((size_t)b * CC + c0) * (HH * WW) + k;
#pragma unroll 4
    for (int c = 0; c < CH_PER_BLK; ++c) {
        const float* xc = xb + (size_t)c * (HH * WW);
        float acc = 0.0f;
#pragma unroll
        for (int s = 0; s < NSAMP; ++s) {
            acc += wgt[s][0] * xc[idx[s][0]];
            acc += wgt[s][1] * xc[idx[s][1]];
            acc += wgt[s][2] * xc[idx[s][2]];
            acc += wgt[s][3] * xc[idx[s][3]];
        }
        sb[(size_t)c * (HH * WW)] = acc * (1.0f / (float)NSAMP);
    }
}

// ---------------------------------------------------------------------------
// Kernel 3a: weight reorder  OIHW -> Wr[tap][ci][co]   (tap = dy*3+dx).
// Makes the WMMA B-fragment loads co-contiguous (coalesced 64B segments).
// ---------------------------------------------------------------------------
__global__ __launch_bounds__(256) void wreorder_kernel(const float* __restrict__ Wt,
                                                       float* __restrict__ Wr) {
    int t = blockIdx.x * blockDim.x + threadIdx.x;
    if (t >= 9 * CC * CC) return;
    int co  = t % CC;
    int ci  = (t / CC) % CC;
    int tap = t / (CC * CC);
    Wr[t] = Wt[(size_t)co * CC * 9 + (size_t)ci * 9 + tap];
}

// ---------------------------------------------------------------------------
// Kernel 3b: 3x3 SAME conv (320->320) + bias + ReLU as implicit GEMM on the
// CDNA5 matrix pipe: V_WMMA_F32_16X16X4_F32 (full fp32, matching reference
// precision).  Tile: 16 output pixels (one row segment) x 16 output channels
// per wave; 4 waves/block -> 16 px x 64 co per block.
//
// Zero-padded 3x18 input patch is cooperatively staged in LDS in 80-channel
// chunks ([ci][3][18], ci-stride 54 floats -> conflict-free ds_loads); all
// four waves reuse it.  B fragments come from the reordered Wr (coalesced).
// Inner loop: 2x ds_load_b32 + 2x coalesced global_load_b32 + 1x v_wmma.
//
// f32 A 16x4 layout:  lanes 0-15 hold M=0..15 with K=0 (v0), K=1 (v1);
//                     lanes 16-31 hold M=0..15 with K=2 (v0), K=3 (v1).
// f32 B 4x16 mirrored: N = lane%16, K = 2*(lane/16) + {0,1}.
// f32 C/D 16x16: VGPR r: lanes 0-15 -> (M=r, N=lane); lanes 16-31 -> (M=r+8).
// ---------------------------------------------------------------------------
__global__ __launch_bounds__(128) void conv_wmma_kernel(const float* __restrict__ S,
                                                        const float* __restrict__ Wr,
                                                        const float* __restrict__ bias,
                                                        float* __restrict__ out) {
    __shared__ float tile[TILE_ELEMS];           // 80 ci * 3 rows * 18 cols = 17.3KB

    const int lane  = threadIdx.x & 31;
    const int wave  = threadIdx.x >> 5;          // 0..3
    const int mlane = lane & 15;
    const int khalf = (lane >> 4) << 1;          // 0 or 2

    const int tileid = blockIdx.x;               // 0..255 = y * (W/16) + xt
    const int y      = tileid >> 2;              // W/16 == 4
    const int x0     = (tileid & 3) << 4;
    const int b      = blockIdx.z;
    const int co0    = blockIdx.y * 64 + wave * 16;

    const float* Sb = S + (size_t)b * CC * (HH * WW);
    const int    co = co0 + mlane;               // this lane's B-matrix column

    v8f acc = {};

    for (int cc0 = 0; cc0 < CC; cc0 += CI_CHUNK) {
        __syncthreads();                         // previous chunk fully consumed
        // ---- stage zero-padded patch: tile[ci][r][col], col=0..17 ----
        for (int i = threadIdx.x; i < TILE_ELEMS; i += 128) {
            int ci  = i / 54;
            int rem = i - ci * 54;
            int r   = rem / 18;
            int col = rem - r * 18;
            int yy = y + r - 1;
            int xx = x0 + col - 1;
            float v = 0.0f;
            if (yy >= 0 && yy < HH && xx >= 0 && xx < WW)
                v = Sb[(size_t)(cc0 + ci) * (HH * WW) + yy * WW + xx];
            tile[i] = v;
        }
        __syncthreads();

        // ---- 9 taps x (CI_CHUNK/4) WMMAs, uniform control flow ----
        for (int dy = 0; dy < 3; ++dy) {
            for (int dx = 0; dx < 3; ++dx) {
                const float* tl = tile + dy * 18 + (mlane + dx);
                const float* wr = Wr + ((size_t)(dy * 3 + dx) * CC + cc0) * CC + co;
#pragma unroll 4
                for (int ci2 = 0; ci2 < CI_CHUNK; ci2 += 4) {
                    v2f a;
                    a.x = tl[(ci2 + khalf)     * 54];
                    a.y = tl[(ci2 + khalf + 1) * 54];
                    v2f bm;
                    bm.x = wr[(size_t)(ci2 + khalf)     * CC];
                    bm.y = wr[(size_t)(ci2 + khalf + 1) * CC];
                    acc = __builtin_amdgcn_wmma_f32_16x16x4_f32(
                        false, a, false, bm, (short)0, acc, false, false);
                }
            }
        }
    }

    // Epilogue: bias + ReLU, two 16B stores per lane (consecutive x).
    const float bv    = bias[co];
    const int   mbase = (lane >> 4) << 3;        // 0 or 8
    float* Ob = out + (((size_t)b * CC + co) * HH + y) * WW + x0 + mbase;
    float4 v0, v1;
    v0.x = fmaxf(acc[0] + bv, 0.0f);
    v0.y = fmaxf(acc[1] + bv, 0.0f);
    v0.z = fmaxf(acc[2] + bv, 0.0f);
    v0.w = fmaxf(acc[3] + bv, 0.0f);
    v1.x = fmaxf(acc[4] + bv, 0.0f);
    v1.y = fmaxf(acc[5] + bv, 0.0f);
    v1.z = fmaxf(acc[6] + bv, 0.0f);
    v1.w = fmaxf(acc[7] + bv, 0.0f);
    *(float4*)(Ob)     = v0;
    *(float4*)(Ob + 4) = v1;
}

// ---------------------------------------------------------------------------
extern "C" void kernel_launch(void* const* d_in, const int* in_sizes, int n_in,
                              void* d_out, int out_size, void* d_ws, size_t ws_size,
                              hipStream_t stream) {
    const float* x  = (const float*)d_in[0];
    const float* Ks = (const float*)d_in[1];
    const float* Kt = (const float*)d_in[2];
    const float* ps = (const float*)d_in[3];
    const float* pt = (const float*)d_in[4];
    const float* cw = (const float*)d_in[5];
    const float* cb = (const float*)d_in[6];
    float* out = (float*)d_out;

    float* G  = (float*)d_ws;                               // 72 floats
    float* S  = (float*)d_ws + 128;                         // B*C*H*W f32 (42MB)
    float* Wr = (float*)d_ws + 128 + (size_t)BB * CC * HH * WW;  // 9*C*C f32 (3.7MB)

    fmat_kernel<<<1, BB, 0, stream>>>(Ks, Kt, ps, pt, G);

    dim3 gs((HH * WW) / 256, CC / CH_PER_BLK, BB);          // 16 x 10 x 8
    sample_kernel<<<gs, 256, 0, stream>>>(x, G, S);

    wreorder_kernel<<<(9 * CC * CC + 255) / 256, 256, 0, stream>>>(cw, Wr);

    dim3 gc((WW / 16) * HH, CC / 64, BB);                   // 256 x 5 x 8
    conv_wmma_kernel<<<gc, 128, 0, stream>>>(S, Wr, cb, out);
}